// MyModel_26963804684787
// MI455X (gfx1250) — compile-verified
//
#include <hip/hip_runtime.h>
#include <hip/hip_bf16.h>

typedef __attribute__((ext_vector_type(2))) float v2f;
typedef __attribute__((ext_vector_type(8))) float v8f;

#define GN 50000
#define GH 64
#define GE 800000

// ---------------------------------------------------------------------------
// Graph preprocessing
// ---------------------------------------------------------------------------
__global__ void k_init_deg(float* __restrict__ deg) {
    int n = blockIdx.x * blockDim.x + threadIdx.x;
    if (n < GN) deg[n] = 1.0f;
}

__global__ void k_zero_misc(int* __restrict__ cnt, float* __restrict__ pooled) {
    int t = threadIdx.x;
    if (t == 0) *cnt = 0;
    if (t < 192) pooled[t] = 0.0f;
}

__global__ void k_degree(const int* __restrict__ ei, const int* __restrict__ nt,
                         float* __restrict__ deg) {
    int e = blockIdx.x * blockDim.x + threadIdx.x;
    if (e >= GE) return;
    int s = ei[e], d = ei[GE + e];
    if (nt[s] == 0 && nt[d] == 0) atomicAdd(&deg[d], 1.0f);
}

// dinv = rsqrt(deg) in place; also count masked nodes (one atomic per wave)
__global__ void k_dinv_count(float* __restrict__ deg, const int* __restrict__ nt,
                             int* __restrict__ cnt) {
    int n = blockIdx.x * blockDim.x + threadIdx.x;
    bool m = false;
    if (n < GN) {
        deg[n] = rsqrtf(deg[n]);
        m = (nt[n] == 0);
    }
    unsigned long long bal = __ballot(m);
    if ((threadIdx.x & 31) == 0) atomicAdd(cnt, (int)__popcll(bal));
}

__global__ void k_coef(const int* __restrict__ ei, const int* __restrict__ nt,
                       const float* __restrict__ dinv, float* __restrict__ coef) {
    int e = blockIdx.x * blockDim.x + threadIdx.x;
    if (e >= GE) return;
    int s = ei[e], d = ei[GE + e];
    float w = (nt[s] == 0 && nt[d] == 0) ? 1.0f : 0.0f;
    coef[e] = w * dinv[s] * dinv[d];
}

__global__ void k_gather_emb(const int* __restrict__ x, const float* __restrict__ emb,
                             float* __restrict__ dst) {
    int t = blockIdx.x * blockDim.x + threadIdx.x;   // N * (H/4) threads
    if (t >= GN * (GH / 4)) return;
    int n = t >> 4, c = t & 15;
    ((float4*)dst)[(size_t)n * 16 + c] =
        ((const float4*)emb)[(size_t)x[n] * 16 + c];
}

// ---------------------------------------------------------------------------
// WMMA fp32 GEMM, one wave = one 16x16 output tile. All dims compile-time:
// store/epilogue addressing folds into 24-bit instruction offsets, K loop
// unrolls x8 (16 iterations for FIN=64), tile decomposition is shift/mask only.
//   EPI==1 (GCN layer):  C  = hw,  C2 = dinv[row]^2 * hw + bias[col]
//   EPI==2 (MLP head):   C  = (relu(hw + cvec[col]))*gamma[col]*bnrs + beta[col]
// RELU_A: relu fused into the A-operand load (prev layer's activation).
// ---------------------------------------------------------------------------
template <int FIN, int FOUT, int EPI, bool RELU_A>
__global__ void k_wmma_gemm(const float* __restrict__ A, const float* __restrict__ W,
                            float* __restrict__ C, float* __restrict__ C2,
                            const int* __restrict__ xidx,
                            const float* __restrict__ dinv,
                            const float* __restrict__ bias,
                            const float* __restrict__ cvec,
                            const float* __restrict__ gamma,
                            const float* __restrict__ beta) {
    constexpr int TILES_N = FOUT / 16;
    const int lane  = threadIdx.x & 31;
    const int wave  = threadIdx.x >> 5;
    const int gid   = blockIdx.x * (blockDim.x >> 5) + wave;
    const int tileM = gid / TILES_N;
    const int tileN = gid % TILES_N;
    if (tileM * 16 >= GN) return;           // uniform per wave: EXEC stays full

    const int half = lane >> 4;             // 0 | 1
    const int l16  = lane & 15;
    const int row  = tileM * 16 + l16;
    const int arow = xidx ? xidx[row] : row;
    const int col  = tileN * 16 + l16;
    const int koff = half * 2;

    const float* __restrict__ Arow = A + (size_t)arow * FIN + koff;
    const float* __restrict__ Wcol = W + (size_t)koff * FOUT + col;

    v8f acc = {};
#pragma unroll 8
    for (int k = 0; k < FIN; k += 4) {
        v2f a, b;
        a.x = Arow[k];
        a.y = Arow[k + 1];
        if (RELU_A) { a.x = fmaxf(a.x, 0.0f); a.y = fmaxf(a.y, 0.0f); }
        b.x = Wcol[(size_t)k * FOUT];
        b.y = Wcol[(size_t)k * FOUT + FOUT];
        acc = __builtin_amdgcn_wmma_f32_16x16x4_f32(
            false, a, false, b, (short)0, acc, false, false);
    }

    const size_t obase = (size_t)(tileM * 16 + half * 8) * FOUT + col;
    if (EPI == 1) {
        const float bb = bias[col];
        const float* __restrict__ dp = dinv + tileM * 16 + half * 8;
        float* __restrict__ cp  = C  + obase;
        float* __restrict__ c2p = C2 + obase;
#pragma unroll
        for (int r = 0; r < 8; ++r) {
            float v  = acc[r];
            float di = dp[r];
            cp [(size_t)r * FOUT] = v;
            c2p[(size_t)r * FOUT] = di * di * v + bb;
        }
    } else {  // EPI == 2
        const float cv = cvec[col];
        const float gs = gamma[col] * rsqrtf(1.0f + 1e-5f);
        const float bt = beta[col];
        float* __restrict__ cp = C + obase;
#pragma unroll
        for (int r = 0; r < 8; ++r)
            cp[(size_t)r * FOUT] = fmaxf(acc[r] + cv, 0.0f) * gs + bt;
    }
}

// ---------------------------------------------------------------------------
// edge messages: out[dst] += coef[e] * hw[src]   (float4 chunk per thread)
// ---------------------------------------------------------------------------
template <int F4>
__global__ void k_edge_msg(const int* __restrict__ ei, const float* __restrict__ coef,
                           const float* __restrict__ hw, float* __restrict__ out) {
    long long t = (long long)blockIdx.x * blockDim.x + threadIdx.x;
    if (t >= (long long)GE * F4) return;
    int e = (int)(t / F4), c = (int)(t % F4);
    float w = coef[e];
    if (w == 0.0f) return;
    int s = ei[e], d = ei[GE + e];
    float4 m = ((const float4*)hw)[(size_t)s * F4 + c];
    float* o = out + (size_t)d * (F4 * 4) + (size_t)c * 4;
    atomicAdd(o + 0, w * m.x);
    atomicAdd(o + 1, w * m.y);
    atomicAdd(o + 2, w * m.z);
    atomicAdd(o + 3, w * m.w);
}

// ---------------------------------------------------------------------------
// masked column-sum of relu(z3): pooled[f] += sum_n (nt[n]==0) relu(z[n,f])
// blockDim = 192 (one thread per column), grid-strided over rows
// ---------------------------------------------------------------------------
__global__ void k_pool(const float* __restrict__ z, const int* __restrict__ nt,
                       float* __restrict__ pooled) {
    int f = threadIdx.x;                    // 0..191
    float s = 0.0f;
    for (int n = blockIdx.x; n < GN; n += gridDim.x)
        if (nt[n] == 0) s += fmaxf(z[(size_t)n * 192 + f], 0.0f);
    atomicAdd(&pooled[f], s);
}

// cvec[j] = bh1[j] + sum_i (pooled[i]/count) * Wh1[i,j]   (top 192 rows of Wh1)
__global__ void k_cvec(const float* __restrict__ pooled, const int* __restrict__ cnt,
                       const float* __restrict__ Wh1, const float* __restrict__ bh1,
                       float* __restrict__ cvec) {
    int j = threadIdx.x;                    // 0..127
    float inv = 1.0f / (float)(*cnt);
    float s = bh1[j];
    for (int i = 0; i < 192; ++i)
        s += pooled[i] * inv * Wh1[(size_t)i * 128 + j];
    cvec[j] = s;
}

// out[n] = dot(h1bn[n,:128], Wh2) + bh2   (one wave per row)
__global__ void k_final(const float* __restrict__ h1, const float* __restrict__ Wh2,
                        const float* __restrict__ bh2, float* __restrict__ out) {
    int lane = threadIdx.x & 31;
    int row  = blockIdx.x * (blockDim.x >> 5) + (threadIdx.x >> 5);
    if (row >= GN) return;
    float4 h = ((const float4*)h1)[(size_t)row * 32 + lane];
    float4 w = ((const float4*)Wh2)[lane];
    float v = h.x * w.x + h.y * w.y + h.z * w.z + h.w * w.w;
#pragma unroll
    for (int off = 16; off > 0; off >>= 1) v += __shfl_down(v, off, 32);
    if (lane == 0) out[row] = v + bh2[0];
}

// ---------------------------------------------------------------------------
extern "C" void kernel_launch(void* const* d_in, const int* in_sizes, int n_in,
                              void* d_out, int out_size, void* d_ws, size_t ws_size,
                              hipStream_t stream) {
    const int*   x    = (const int*)d_in[0];
    const int*   ei   = (const int*)d_in[1];   // [2,E] flat
    const int*   nt   = (const int*)d_in[2];
    const float* emb  = (const float*)d_in[3];
    const float* W1   = (const float*)d_in[4];
    const float* b1   = (const float*)d_in[5];
    const float* W2   = (const float*)d_in[6];
    const float* b2   = (const float*)d_in[7];
    const float* W3   = (const float*)d_in[8];
    const float* b3   = (const float*)d_in[9];
    const float* Wh1  = (const float*)d_in[10];
    const float* bh1  = (const float*)d_in[11];
    const float* gamma= (const float*)d_in[12];
    const float* beta = (const float*)d_in[13];
    const float* Wh2  = (const float*)d_in[14];
    const float* bh2  = (const float*)d_in[15];
    float* out = (float*)d_out;

    // workspace carve-up (256B aligned)
    size_t off = 0;
    auto alloc = [&](size_t bytes) {
        size_t p = off;
        off = (off + bytes + 255) & ~(size_t)255;
        return (char*)d_ws + p;
    };
    float* dinv   = (float*)alloc((size_t)GN * 4);      // deg -> dinv in place
    float* coef   = (float*)alloc((size_t)GE * 4);
    int*   cnt    = (int*)  alloc(256);
    float* pooled = (float*)alloc(192 * 4);
    float* cvec   = (float*)alloc(128 * 4);
    float* bufA   = (float*)alloc((size_t)GN * 256 * 4);
    float* bufHW  = (float*)alloc((size_t)GN * 256 * 4);
    float* bufOUT = (float*)alloc((size_t)GN * 256 * 4);
    (void)ws_size;

    const int TB = 256;
    auto blocks = [&](long long total) { return (unsigned)((total + TB - 1) / TB); };
    auto gemm_blocks = [&](int fout) {
        return blocks((long long)(GN / 16) * (fout / 16) * 32);
    };

    // --- graph preprocessing ---
    k_init_deg<<<blocks(GN), TB, 0, stream>>>(dinv);
    k_zero_misc<<<1, 256, 0, stream>>>(cnt, pooled);
    k_degree<<<blocks(GE), TB, 0, stream>>>(ei, nt, dinv);
    k_dinv_count<<<blocks(GN), TB, 0, stream>>>(dinv, nt, cnt);
    k_coef<<<blocks(GE), TB, 0, stream>>>(ei, nt, dinv, coef);
    k_gather_emb<<<blocks((long long)GN * 16), TB, 0, stream>>>(x, emb, bufA);

    // --- layer 1: z0(64) -> hw(256); z1 = agg in bufOUT ---
    k_wmma_gemm<64, 256, 1, false><<<gemm_blocks(256), TB, 0, stream>>>(
        bufA, W1, bufHW, bufOUT, nullptr, dinv, b1, nullptr, nullptr, nullptr);
    k_edge_msg<64><<<blocks((long long)GE * 64), TB, 0, stream>>>(
        ei, coef, bufHW, bufOUT);

    // --- layer 2: relu(z1)(256) -> hw(256); z2 in bufA ---
    k_wmma_gemm<256, 256, 1, true><<<gemm_blocks(256), TB, 0, stream>>>(
        bufOUT, W2, bufHW, bufA, nullptr, dinv, b2, nullptr, nullptr, nullptr);
    k_edge_msg<64><<<blocks((long long)GE * 64), TB, 0, stream>>>(
        ei, coef, bufHW, bufA);

    // --- layer 3: relu(z2)(256) -> hw(192); z3 in bufOUT ---
    k_wmma_gemm<256, 192, 1, true><<<gemm_blocks(192), TB, 0, stream>>>(
        bufA, W3, bufHW, bufOUT, nullptr, dinv, b3, nullptr, nullptr, nullptr);
    k_edge_msg<48><<<blocks((long long)GE * 48), TB, 0, stream>>>(
        ei, coef, bufHW, bufOUT);

    // --- masked mean pool over relu(z3) ---
    k_pool<<<256, 192, 0, stream>>>(bufOUT, nt, pooled);
    k_cvec<<<1, 128, 0, stream>>>(pooled, cnt, Wh1, bh1, cvec);

    // --- head: h1 = BN(relu(emb @ Wh1[192:,:] + cvec)) via WMMA epilogue ---
    k_wmma_gemm<64, 128, 2, false><<<gemm_blocks(128), TB, 0, stream>>>(
        emb, Wh1 + (size_t)192 * 128, bufHW, nullptr, x,
        nullptr, nullptr, cvec, gamma, beta);

    // --- final 128 -> 1 projection, one wave per row ---
    k_final<<<blocks((long long)GN * 4), TB, 0, stream>>>(bufHW, Wh2, bh2, out);
}